// StateNetwork_11416023073407
// MI455X (gfx1250) — compile-verified
//
#include <hip/hip_runtime.h>
#include <hip/hip_bf16.h>
#include <stdint.h>

// Problem constants (match reference)
#define NN    1024
#define BB    32
#define FIN   25
#define FOUT  3
#define NG    5
#define FCOUT 50
#define KFC   (NN * FOUT)   // 3072

typedef float v2f __attribute__((ext_vector_type(2)));
typedef float v8f __attribute__((ext_vector_type(8)));

__device__ __forceinline__ float lrelu(float v) { return v > 0.0f ? v : 0.2f * v; }

// Workspace layout (float units)
//  whf2  : [NG][NN] float4 (wh0,wh1,wh2,f2)   off 0       size 20480
//  f1    : [NG][NN]                           off 20480   size 5120
//  maxF2 : [NG] (padded)                      off 25600   size 8
//  meanWh: [NG][4]                            off 25608   size 24 (pad to 32)
//  outs  : [NG][BB][NN][FOUT]                 off 25632   size 491520
// total ~2.07 MB
#define WS_WHF2   0
#define WS_F1     20480
#define WS_MAXF2  25600
#define WS_MEANWH 25608
#define WS_OUTS   25632

// ---------------------------------------------------------------------------
// Phase 1: Wh = x@W, f1/f2 projections, max(f2) and mean(Wh) per branch.
// One block per branch, one thread per entity row.
// ---------------------------------------------------------------------------
__global__ __launch_bounds__(1024) void gat_prep(const float* __restrict__ x,
                                                 const float* __restrict__ W,
                                                 const float* __restrict__ a,
                                                 float* __restrict__ ws) {
    const int g = blockIdx.x;
    const int i = threadIdx.x;

    __shared__ float  sW[FIN * FOUT];
    __shared__ float  sa[2 * FOUT];
    __shared__ float4 red[NN];

    if (i < FIN * FOUT) sW[i] = W[g * FIN * FOUT + i];
    if (i >= FIN * FOUT && i < FIN * FOUT + 2 * FOUT)
        sa[i - FIN * FOUT] = a[g * 2 * FOUT + (i - FIN * FOUT)];
    __syncthreads();

    float wh0 = 0.f, wh1 = 0.f, wh2 = 0.f;
    const float* xr = x + i * FIN;
#pragma unroll
    for (int c = 0; c < FIN; ++c) {
        float xv = xr[c];
        wh0 += xv * sW[c * 3 + 0];
        wh1 += xv * sW[c * 3 + 1];
        wh2 += xv * sW[c * 3 + 2];
    }
    float f1 = wh0 * sa[0] + wh1 * sa[1] + wh2 * sa[2];
    float f2 = wh0 * sa[3] + wh1 * sa[4] + wh2 * sa[5];

    float4 pack;
    pack.x = wh0; pack.y = wh1; pack.z = wh2; pack.w = f2;
    ((float4*)(ws + WS_WHF2))[g * NN + i] = pack;
    ws[WS_F1 + g * NN + i] = f1;

    // block reduction: sum wh (xyz), max f2 (w)
    red[i] = pack;
    __syncthreads();
    for (int s = NN / 2; s > 0; s >>= 1) {
        if (i < s) {
            float4 o = red[i + s];
            red[i].x += o.x; red[i].y += o.y; red[i].z += o.z;
            red[i].w = fmaxf(red[i].w, o.w);
        }
        __syncthreads();
    }
    if (i == 0) {
        ws[WS_MAXF2 + g]          = red[0].w;
        ws[WS_MEANWH + g * 4 + 0] = red[0].x * (1.0f / NN);
        ws[WS_MEANWH + g * 4 + 1] = red[0].y * (1.0f / NN);
        ws[WS_MEANWH + g * 4 + 2] = red[0].z * (1.0f / NN);
        ws[WS_MEANWH + g * 4 + 3] = 0.f;
    }
}

// ---------------------------------------------------------------------------
// Phase 2: fused masked-softmax + att@Wh + ELU. One wave32 per output row.
// Streams the 168MB adj tensor exactly once, 128B coalesced per wave-iter.
// ---------------------------------------------------------------------------
__global__ __launch_bounds__(256) void gat_rows(const uint8_t* __restrict__ adj,
                                                const float* __restrict__ wsc,
                                                float* __restrict__ outs) {
    const int wave = threadIdx.x >> 5;
    const int lane = threadIdx.x & 31;

    // grid: NG * BB * (NN/8) blocks; 8 waves per block -> 8 rows per block
    const int blk = blockIdx.x;
    const int g   = blk / (BB * (NN / 8));
    const int rem = blk % (BB * (NN / 8));
    const int b   = rem / (NN / 8);
    const int i   = (rem % (NN / 8)) * 8 + wave;

    const uint8_t* arow = adj + (((size_t)g * BB + b) * NN + i) * NN;
    __builtin_prefetch(arow + NN, 0, 1);   // -> global_prefetch_b8, warm next row

    const float4* whg = ((const float4*)(wsc + WS_WHF2)) + g * NN;
    const float   f1i = wsc[WS_F1 + g * NN + i];
    const float   Mi  = lrelu(f1i + wsc[WS_MAXF2 + g]);  // safe softmax shift

    float a0 = 0.f, a1 = 0.f, a2 = 0.f, ad = 0.f;
    int   cnt = 0;

#pragma unroll
    for (int it = 0; it < 8; ++it) {
        const int j0 = it * 128 + lane * 4;
        uchar4 m = *(const uchar4*)(arow + j0);
        cnt += (int)m.x + (int)m.y + (int)m.z + (int)m.w;
        float mk[4] = { (float)m.x, (float)m.y, (float)m.z, (float)m.w };
#pragma unroll
        for (int k = 0; k < 4; ++k) {
            float4 w = whg[j0 + k];                        // (wh0,wh1,wh2,f2)
            float  e = __expf(lrelu(f1i + w.w) - Mi) * mk[k];
            a0 += e * w.x; a1 += e * w.y; a2 += e * w.z; ad += e;
        }
    }

    // wave32 butterfly reduction
    for (int off = 16; off > 0; off >>= 1) {
        a0  += __shfl_xor(a0, off);
        a1  += __shfl_xor(a1, off);
        a2  += __shfl_xor(a2, off);
        ad  += __shfl_xor(ad, off);
        cnt += __shfl_xor(cnt, off);
    }

    if (lane == 0) {
        float o0, o1, o2;
        if (cnt == 0) {  // all-masked row: softmax of equal -9e15 logits = uniform
            o0 = wsc[WS_MEANWH + g * 4 + 0];
            o1 = wsc[WS_MEANWH + g * 4 + 1];
            o2 = wsc[WS_MEANWH + g * 4 + 2];
        } else {
            float inv = 1.0f / ad;
            o0 = a0 * inv; o1 = a1 * inv; o2 = a2 * inv;
        }
        // ELU
        o0 = o0 > 0.f ? o0 : __expf(o0) - 1.0f;
        o1 = o1 > 0.f ? o1 : __expf(o1) - 1.0f;
        o2 = o2 > 0.f ? o2 : __expf(o2) - 1.0f;
        float* dst = outs + (((size_t)g * BB + b) * NN + i) * FOUT;
        dst[0] = o0; dst[1] = o1; dst[2] = o2;
    }
}

// ---------------------------------------------------------------------------
// Phase 3: FC per branch: C(32x50) = A(32x3072) * fc_w^T + fc_b via
// V_WMMA_F32_16X16X4_F32.
//  - Invalid output columns (50..63) load a CLAMPED valid fc_w row
//    unconditionally: D column n depends only on B column n, and those
//    columns are never stored. No exec-mask churn in the inner loop.
//  - 4 independent accumulators break the D->C WMMA dependency chain.
// ---------------------------------------------------------------------------
#if __has_builtin(__builtin_amdgcn_wmma_f32_16x16x4_f32)

__device__ __forceinline__ v8f wmma4(v2f av, v2f bv, v8f c) {
    return __builtin_amdgcn_wmma_f32_16x16x4_f32(
        /*neg_a=*/false, av, /*neg_b=*/false, bv,
        /*c_mod=*/(short)0, c, /*reuse_a=*/false, /*reuse_b=*/false);
}

__global__ __launch_bounds__(32) void fc_wmma(const float* __restrict__ outs,
                                              const float* __restrict__ fcw,
                                              const float* __restrict__ fcb,
                                              float* __restrict__ out) {
    const int g    = blockIdx.x >> 3;
    const int t    = blockIdx.x & 7;
    const int mt   = t >> 2;          // 0..1  (rows b: 0-15 / 16-31)
    const int nt   = t & 3;           // 0..3  (cols o: 0-15/16-31/32-47/48-63)
    const int lane = threadIdx.x;
    const int half = lane >> 4;       // K-half select: 0 -> K{0,1}, 1 -> K{2,3}
    const int lidx = lane & 15;
    const int koff = half * 2;

    // A frag per ISA layout: lane(l<16) holds A[M=l][K=0..1], lane(l>=16) A[M=l-16][K=2..3]
    const float* Arow = outs + (size_t)g * BB * KFC + (size_t)(mt * 16 + lidx) * KFC;
    // B frag mirrored, B[k][n] = fc_w[g][o=n][k]; clamp o so every lane loads
    // a real row (values for o>=50 land in D columns that are never stored).
    const int ocol  = nt * 16 + lidx;
    const int ocolc = ocol < FCOUT ? ocol : (FCOUT - 1);
    const float* Brow = fcw + (size_t)g * FCOUT * KFC + (size_t)ocolc * KFC;

    v8f c0 = {0.f,0.f,0.f,0.f,0.f,0.f,0.f,0.f};
    v8f c1 = c0, c2 = c0, c3 = c0;

#pragma unroll 2
    for (int k = 0; k < KFC; k += 16) {
        v2f a0v = *(const v2f*)(Arow + k      + koff);
        v2f b0v = *(const v2f*)(Brow + k      + koff);
        v2f a1v = *(const v2f*)(Arow + k + 4  + koff);
        v2f b1v = *(const v2f*)(Brow + k + 4  + koff);
        v2f a2v = *(const v2f*)(Arow + k + 8  + koff);
        v2f b2v = *(const v2f*)(Brow + k + 8  + koff);
        v2f a3v = *(const v2f*)(Arow + k + 12 + koff);
        v2f b3v = *(const v2f*)(Brow + k + 12 + koff);
        c0 = wmma4(a0v, b0v, c0);
        c1 = wmma4(a1v, b1v, c1);
        c2 = wmma4(a2v, b2v, c2);
        c3 = wmma4(a3v, b3v, c3);
    }
    v8f c = (c0 + c1) + (c2 + c3);

    if (ocol < FCOUT) {
        float bias = fcb[g * FCOUT + ocol];
#pragma unroll
        for (int r = 0; r < 8; ++r) {
            int bb = mt * 16 + (half ? (8 + r) : r);   // D: VGPR r -> M=r / M=r+8
            out[g * BB * FCOUT + bb * FCOUT + ocol] = c[r] + bias;
        }
    }
}
#else
// Fallback: scalar FC, one thread per (g,b,o).
__global__ __launch_bounds__(256) void fc_wmma(const float* __restrict__ outs,
                                               const float* __restrict__ fcw,
                                               const float* __restrict__ fcb,
                                               float* __restrict__ out) {
    int idx = blockIdx.x * blockDim.x + threadIdx.x;
    if (idx >= NG * BB * FCOUT) return;
    int g = idx / (BB * FCOUT);
    int r = idx % (BB * FCOUT);
    int b = r / FCOUT, o = r % FCOUT;
    const float* A = outs + (size_t)g * BB * KFC + (size_t)b * KFC;
    const float* Wv = fcw + (size_t)g * FCOUT * KFC + (size_t)o * KFC;
    float acc = fcb[g * FCOUT + o];
    for (int k = 0; k < KFC; ++k) acc += A[k] * Wv[k];
    out[idx] = acc;
}
#endif

extern "C" void kernel_launch(void* const* d_in, const int* in_sizes, int n_in,
                              void* d_out, int out_size, void* d_ws, size_t ws_size,
                              hipStream_t stream) {
    const float*   x   = (const float*)d_in[0];
    const uint8_t* adj = (const uint8_t*)d_in[1];   // numpy bool_ -> 1 byte/element
    const float*   W   = (const float*)d_in[2];
    const float*   a   = (const float*)d_in[3];
    const float*   fcw = (const float*)d_in[4];
    const float*   fcb = (const float*)d_in[5];
    float* out  = (float*)d_out;
    float* ws   = (float*)d_ws;
    float* outs = ws + WS_OUTS;

    gat_prep<<<NG, 1024, 0, stream>>>(x, W, a, ws);
    gat_rows<<<NG * BB * (NN / 8), 256, 0, stream>>>(adj, ws, outs);
#if __has_builtin(__builtin_amdgcn_wmma_f32_16x16x4_f32)
    fc_wmma<<<NG * 8, 32, 0, stream>>>(outs, fcw, fcb, out);
#else
    fc_wmma<<<(NG * BB * FCOUT + 255) / 256, 256, 0, stream>>>(outs, fcw, fcb, out);
#endif
}